// GNN_22505628631761
// MI455X (gfx1250) — compile-verified
//
#include <hip/hip_runtime.h>
#include <hip/hip_bf16.h>
#include <math.h>

typedef float v2f __attribute__((ext_vector_type(2)));
typedef float v8f __attribute__((ext_vector_type(8)));

#define HIDW 128

// ---------------------------------------------------------------------------
// degree kernels
// ---------------------------------------------------------------------------
__global__ void deg_init_kernel(float* __restrict__ deg, int n) {
    int i = blockIdx.x * blockDim.x + threadIdx.x;
    if (i < n) deg[i] = 1.0f;  // self loop
}

__global__ void deg_count_kernel(const int* __restrict__ dst, float* __restrict__ deg, int e) {
    int i = blockIdx.x * blockDim.x + threadIdx.x;
    if (i < e) atomicAdd(&deg[dst[i]], 1.0f);
}

__global__ void deg_rsqrt_kernel(float* __restrict__ deg, int n) {
    int i = blockIdx.x * blockDim.x + threadIdx.x;
    if (i < n) deg[i] = rsqrtf(deg[i]);
}

// ---------------------------------------------------------------------------
// WMMA GEMM: C[Mx128] = A[MxK] @ B[Kx128]. Fully templated so the inner loop
// is branch-free. Each wave computes TWO 16x16 output tiles (rows base and
// base+16, same 16-col tile) sharing B fragments: per k-step that is
// 2 x global_load_b64 (A) + 2 x global_load_b32 (B) feeding 2 WMMAs.
// Block = 256 threads = 8 waves -> 32 rows x 128 cols per block.
// Optional: second B matrix (DUAL, shares A fragments; out1 gets bias1),
// bias on out0 (BIAS0), per-row dinv scale (SCALE), and an extra store of
// value*dinv^2 (AGG, the GCN self-loop term).
// Safe in place (out0 == A): __syncthreads separates the K-loop reads from
// the stores, and the 32 A rows are block-private.
// ---------------------------------------------------------------------------
template <int K, bool DUAL, bool BIAS0, bool SCALE, bool AGG>
__global__ void __launch_bounds__(256)
wmma_gemm128_kernel(const float* A,
                    const float* __restrict__ B0, const float* __restrict__ bias0,
                    float* out0,
                    const float* __restrict__ B1, const float* __restrict__ bias1,
                    float* __restrict__ out1,
                    float* __restrict__ agg0, const float* __restrict__ dinv)
{
    const int wave = threadIdx.x >> 5;
    const int lane = threadIdx.x & 31;
    const int half = lane >> 4;    // 0: lanes 0-15, 1: lanes 16-31
    const int l16  = lane & 15;
    const int col  = wave * 16 + l16;                    // output column (0..127)
    const size_t rowBase = (size_t)blockIdx.x * 32;
    const float* Ap0 = A + (rowBase + l16) * (size_t)K;  // row tile 0
    const float* Ap1 = Ap0 + (size_t)16 * K;             // row tile 1

    v8f acc0[2] = {};   // B0 accumulators, tiles 0/1
    v8f acc1[2] = {};   // B1 accumulators (DUAL)

    #pragma unroll 8
    for (int k = 0; k < K; k += 4) {
        const int kk = k + half * 2;  // lane half selects K pair
        // A 16x4 fragments (two row tiles share the B fragments)
        v2f a0 = *(const v2f*)(Ap0 + kk);
        v2f a1 = *(const v2f*)(Ap1 + kk);
        // B 4x16 fragment: per lane (B[kk][col], B[kk+1][col])
        v2f b;
        b.x = B0[(size_t)(kk)     * HIDW + col];
        b.y = B0[(size_t)(kk + 1) * HIDW + col];
        acc0[0] = __builtin_amdgcn_wmma_f32_16x16x4_f32(false, a0, false, b,
                                                        (short)0, acc0[0], false, false);
        acc0[1] = __builtin_amdgcn_wmma_f32_16x16x4_f32(false, a1, false, b,
                                                        (short)0, acc0[1], false, false);
        if (DUAL) {
            v2f b1v;
            b1v.x = B1[(size_t)(kk)     * HIDW + col];
            b1v.y = B1[(size_t)(kk + 1) * HIDW + col];
            acc1[0] = __builtin_amdgcn_wmma_f32_16x16x4_f32(false, a0, false, b1v,
                                                            (short)0, acc1[0], false, false);
            acc1[1] = __builtin_amdgcn_wmma_f32_16x16x4_f32(false, a1, false, b1v,
                                                            (short)0, acc1[1], false, false);
        }
    }

    __syncthreads();  // all waves done reading A rows before in-place stores

    // C/D layout: VGPR r holds M = r (lanes 0-15) and M = r+8 (lanes 16-31)
    #pragma unroll
    for (int t = 0; t < 2; ++t) {
        #pragma unroll
        for (int r = 0; r < 8; ++r) {
            const size_t row = rowBase + t * 16 + r + half * 8;
            float v = acc0[t][r];
            if (BIAS0) v += bias0[col];
            if (SCALE) {
                const float di = dinv[row];
                v *= di;                                   // hs = (A@B) * dinv
                if (AGG) agg0[row * HIDW + col] = v * di;  // self loop: h * dinv^2
            }
            out0[row * HIDW + col] = v;
            if (DUAL) {
                out1[row * HIDW + col] = acc1[t][r] + bias1[col];
            }
        }
    }
}

// ---------------------------------------------------------------------------
// Edge aggregation: one wave per edge, 32 lanes x float4 = 128 features.
// agg[dst] += hs[src] * dinv[dst]   (hs is already scaled by dinv[src])
// h / agg are L2-resident (51MB each, 192MB L2) so atomics stay on-chip.
// ---------------------------------------------------------------------------
__global__ void __launch_bounds__(256)
edge_scatter_kernel(const float* __restrict__ hs, float* __restrict__ agg,
                    const int* __restrict__ src, const int* __restrict__ dst,
                    const float* __restrict__ dinv, int e)
{
    const int edge = blockIdx.x * (blockDim.x >> 5) + (threadIdx.x >> 5);
    if (edge >= e) return;
    const int lane = threadIdx.x & 31;
    const int s = src[edge];
    const int d = dst[edge];
    const float coef = dinv[d];
    const float4 v = *((const float4*)(hs + (size_t)s * HIDW) + lane);
    float* ap = agg + (size_t)d * HIDW + lane * 4;
    atomicAdd(ap + 0, v.x * coef);
    atomicAdd(ap + 1, v.y * coef);
    atomicAdd(ap + 2, v.z * coef);
    atomicAdd(ap + 3, v.w * coef);
}

// ---------------------------------------------------------------------------
// elementwise: a = relu(a + bias[col])
// ---------------------------------------------------------------------------
__global__ void bias_relu_kernel(float* __restrict__ a, const float* __restrict__ bias,
                                 size_t total)
{
    size_t i = (size_t)blockIdx.x * blockDim.x + threadIdx.x;
    if (i >= total) return;
    float v = a[i] + bias[i & (HIDW - 1)];
    a[i] = v > 0.0f ? v : 0.0f;
}

// elementwise: out = agg + bias[col] + orig   (residual)
__global__ void residual_kernel(const float* __restrict__ agg, const float* __restrict__ bias,
                                const float* __restrict__ orig, float* __restrict__ out,
                                size_t total)
{
    size_t i = (size_t)blockIdx.x * blockDim.x + threadIdx.x;
    if (i >= total) return;
    out[i] = agg[i] + bias[i & (HIDW - 1)] + orig[i];
}

// ---------------------------------------------------------------------------
// Final heads: one wave per node (wave32). log_softmax over 128 feats and
// the scalar degree head, via __shfl_xor reductions.
// ---------------------------------------------------------------------------
__global__ void __launch_bounds__(256)
finalize_kernel(const float* __restrict__ h, const float* __restrict__ wdeg,
                const float* __restrict__ bdeg,
                float* __restrict__ res1, float* __restrict__ res2, int n)
{
    const int node = blockIdx.x * (blockDim.x >> 5) + (threadIdx.x >> 5);
    if (node >= n) return;
    const int lane = threadIdx.x & 31;
    const float4 hv = *((const float4*)(h + (size_t)node * HIDW) + lane);

    // row max
    float m = fmaxf(fmaxf(hv.x, hv.y), fmaxf(hv.z, hv.w));
    for (int off = 16; off > 0; off >>= 1) m = fmaxf(m, __shfl_xor(m, off, 32));
    // sum of exp
    float s = __expf(hv.x - m) + __expf(hv.y - m) + __expf(hv.z - m) + __expf(hv.w - m);
    for (int off = 16; off > 0; off >>= 1) s += __shfl_xor(s, off, 32);
    const float lse = m + __logf(s);

    float4 o;
    o.x = hv.x - lse; o.y = hv.y - lse; o.z = hv.z - lse; o.w = hv.w - lse;
    *((float4*)(res1 + (size_t)node * HIDW) + lane) = o;

    // degree head: dot(h, Wdeg) + bdeg
    const float4 wv = *((const float4*)wdeg + lane);
    float d = hv.x * wv.x + hv.y * wv.y + hv.z * wv.z + hv.w * wv.w;
    for (int off = 16; off > 0; off >>= 1) d += __shfl_xor(d, off, 32);
    if (lane == 0) res2[node] = d + bdeg[0];
}

// ---------------------------------------------------------------------------
// launcher
// ---------------------------------------------------------------------------
extern "C" void kernel_launch(void* const* d_in, const int* in_sizes, int n_in,
                              void* d_out, int out_size, void* d_ws, size_t ws_size,
                              hipStream_t stream) {
    const float* x    = (const float*)d_in[0];
    const int*   ei   = (const int*)  d_in[1];
    const float* W1   = (const float*)d_in[2];
    const float* b1   = (const float*)d_in[3];
    const float* W2   = (const float*)d_in[4];
    const float* b2   = (const float*)d_in[5];
    const float* Wd   = (const float*)d_in[6];
    const float* bd   = (const float*)d_in[7];
    const float* Wdeg = (const float*)d_in[8];
    const float* bdeg = (const float*)d_in[9];
    const float* W3   = (const float*)d_in[10];
    const float* b3   = (const float*)d_in[11];

    const int D = 256;                     // input feature width
    const int N = in_sizes[0] / D;         // 100000
    const int E = in_sizes[1] / 2;         // 1600000
    const int* src = ei;
    const int* dst = ei + E;

    // outputs, concatenated flat: res1 [N,128], res2 [N], res3 [N,128]
    float* res1 = (float*)d_out;
    float* res2 = res1 + (size_t)N * HIDW;
    float* res3 = res2 + (size_t)N;

    // workspace: dinv | bufA | bufB | bufC  (~154 MB)
    float* dinv = (float*)d_ws;
    float* bufA = dinv + (((size_t)N + 255) / 256) * 256;  // hs1 -> agg2
    float* bufB = bufA + (size_t)N * HIDW;                 // orig (residual)
    float* bufC = bufB + (size_t)N * HIDW;                 // agg1 -> h1r -> hs2 -> h

    const size_t total = (size_t)N * HIDW;
    const int TB = 256;
    const int gN   = (N + TB - 1) / TB;
    const int gE   = (E + TB - 1) / TB;
    const int gTot = (int)((total + TB - 1) / TB);
    const int gEdgeWaves = (E + 7) / 8;    // 8 waves per 256-thread block
    const int gNodeWaves = (N + 7) / 8;
    const int gGemm = N / 32;              // 100000 / 32 = 3125 (exact)

    // 1) symmetric-norm coefficients
    deg_init_kernel<<<gN, TB, 0, stream>>>(dinv, N);
    deg_count_kernel<<<gE, TB, 0, stream>>>(dst, dinv, E);
    deg_rsqrt_kernel<<<gN, TB, 0, stream>>>(dinv, N);

    // 2) fused GEMM1: hs1 = (x@W1)*dinv, agg1 = hs1*dinv, orig = x@Wd + bd
    wmma_gemm128_kernel<256, true, false, true, true><<<gGemm, TB, 0, stream>>>(
        x, W1, nullptr, bufA, Wd, bd, bufB, bufC, dinv);

    // 3) conv1 edge aggregation: agg1[dst] += hs1[src]*dinv[dst]
    edge_scatter_kernel<<<gEdgeWaves, TB, 0, stream>>>(bufA, bufC, src, dst, dinv, E);

    // 4) h1r = relu(agg1 + b1) in place
    bias_relu_kernel<<<gTot, TB, 0, stream>>>(bufC, b1, total);

    // 5) GEMM2 in place: hs2 = (h1r@W2)*dinv (bufC), agg2 = hs2*dinv (bufA)
    wmma_gemm128_kernel<128, false, false, true, true><<<gGemm, TB, 0, stream>>>(
        bufC, W2, nullptr, bufC, nullptr, nullptr, nullptr, bufA, dinv);

    // 6) conv2 edge aggregation: agg2[dst] += hs2[src]*dinv[dst]
    edge_scatter_kernel<<<gEdgeWaves, TB, 0, stream>>>(bufC, bufA, src, dst, dinv, E);

    // 7) residual: h = agg2 + b2 + orig  (into bufC)
    residual_kernel<<<gTot, TB, 0, stream>>>(bufA, b2, bufB, bufC, total);

    // 8) result3 = h @ W3 + b3 (straight WMMA GEMM into d_out)
    wmma_gemm128_kernel<128, false, true, false, false><<<gGemm, TB, 0, stream>>>(
        bufC, W3, b3, res3, nullptr, nullptr, nullptr, nullptr, nullptr);

    // 9) result1 = log_softmax(h), result2 = h @ Wdeg + bdeg
    finalize_kernel<<<gNodeWaves, TB, 0, stream>>>(bufC, Wdeg, bdeg, res1, res2, N);
}